// SpareseResults_40166534152891
// MI455X (gfx1250) — compile-verified
//
#include <hip/hip_runtime.h>

// Stream compaction: per row of a [4096 x 30522] fp32 matrix, emit the first
// 512 nonzeros as (col_as_float, value) into [B, 2, 512], zero-padded.
// Memory-bound: ~500 MB single-use read -> ~22 us floor @ 23.3 TB/s.
// One wave32 per row; ballot+popcount gives exact column-ordered ranks with no
// LDS/barriers. NT loads (th:TH_LOAD_NT) keep the single-use stream out of the
// 192 MB L2. 256 elems per iteration = FOUR independent b64 loads per lane in
// flight (1 KB outstanding per wave) to cover HBM latency at the modest wave
// count (4096 waves = one per row).

typedef __attribute__((ext_vector_type(2))) float v2f;

#define V_COLS 30522
#define PAD_N  512
#define ROWS_PER_BLOCK 8   // 8 waves of 32 per 256-thread block
#define UNROLL 4           // 4 x (32 lanes x float2) = 256 elems per trip

__global__ __launch_bounds__(256) void sparse_pack_kernel(
    const float* __restrict__ in, float* __restrict__ out, int nrows) {
  const int lane   = threadIdx.x & 31;
  const int waveId = threadIdx.x >> 5;
  const int row    = blockIdx.x * ROWS_PER_BLOCK + waveId;
  if (row >= nrows) return;  // wave-uniform

  const float* __restrict__ rowp = in + (size_t)row * V_COLS;
  float* __restrict__ oIdx = out + (size_t)row * (2 * PAD_N);  // plane 0: indices
  float* __restrict__ oVal = oIdx + PAD_N;                     // plane 1: values

  const unsigned lt = (1u << lane) - 1u;  // lanes strictly below me
  unsigned base = 0;                      // wave-uniform running nonzero count

  constexpr int CHUNK = 64 * UNROLL;                     // 256
  constexpr int nFullElems = (V_COLS / CHUNK) * CHUNK;   // 30464

  for (int c = 0; c < nFullElems; c += CHUNK) {
    // Issue all 4 NT loads before any consumer -> 4 outstanding b64 loads.
    // 8-byte aligned everywhere: row stride 122088 B and (c + 64u + 2*lane)*4
    // are both multiples of 8 (b128 would be misaligned on odd rows).
    v2f v[UNROLL];
#pragma unroll
    for (int u = 0; u < UNROLL; ++u) {
      v[u] = __builtin_nontemporal_load(
          reinterpret_cast<const v2f*>(rowp + c + 64 * u + 2 * lane));
    }

    unsigned sub = 0;  // nonzeros in earlier 64-element sub-chunks (uniform)
#pragma unroll
    for (int u = 0; u < UNROLL; ++u) {
      const int colBase = c + 64 * u + 2 * lane;
      const bool nz0 = (v[u].x != 0.0f);
      const bool nz1 = (v[u].y != 0.0f);
      const unsigned m0 = (unsigned)__ballot(nz0);
      const unsigned m1 = (unsigned)__ballot(nz1);
      // Column order within a sub-chunk is lane-major (col = colBase + j),
      // and every earlier sub-chunk wholly precedes this one.
      const unsigned r0 = base + sub +
          (unsigned)__popc(m0 & lt) + (unsigned)__popc(m1 & lt);
      if (nz0 && r0 < PAD_N) {
        oIdx[r0] = (float)colBase;
        oVal[r0] = v[u].x;
      }
      const unsigned r1 = r0 + (nz0 ? 1u : 0u);
      if (nz1 && r1 < PAD_N) {
        oIdx[r1] = (float)(colBase + 1);
        oVal[r1] = v[u].y;
      }
      sub += (unsigned)__popc(m0) + (unsigned)__popc(m1);
    }
    base += sub;
    // No early exit: with ~384 nnz/row it never triggers, and its absence lets
    // the compiler pipeline the loads. Ranks >= PAD_N are simply not stored.
  }

  // Scalar tail for the last V_COLS - 30464 = 58 columns.
  for (int cc = nFullElems; cc < V_COLS; cc += 32) {
    const int col = cc + lane;
    float v = 0.0f;
    if (col < V_COLS) v = __builtin_nontemporal_load(rowp + col);
    const bool nz = (v != 0.0f);
    const unsigned m = (unsigned)__ballot(nz);
    const unsigned r = base + (unsigned)__popc(m & lt);
    if (nz && r < PAD_N) {
      oIdx[r] = (float)col;
      oVal[r] = v;
    }
    base += (unsigned)__popc(m);
  }

  // Zero the padded tail in both planes: harness poisons d_out with 0xAA and
  // re-validates after graph replays, so every element must be written.
  const unsigned count = base < PAD_N ? base : (unsigned)PAD_N;
  for (unsigned k = count + (unsigned)lane; k < PAD_N; k += 32) {
    oIdx[k] = 0.0f;
    oVal[k] = 0.0f;
  }
}

extern "C" void kernel_launch(void* const* d_in, const int* in_sizes, int n_in,
                              void* d_out, int out_size, void* d_ws, size_t ws_size,
                              hipStream_t stream) {
  (void)n_in; (void)out_size; (void)d_ws; (void)ws_size;
  const float* in = (const float*)d_in[0];
  float* out = (float*)d_out;
  const int nrows = in_sizes[0] / V_COLS;                             // 4096
  const int nblocks = (nrows + ROWS_PER_BLOCK - 1) / ROWS_PER_BLOCK;  // 512
  sparse_pack_kernel<<<nblocks, 256, 0, stream>>>(in, out, nrows);
}